// SqExBlock_19954418057527
// MI455X (gfx1250) — compile-verified
//
#include <hip/hip_runtime.h>
#include <math.h>

typedef __attribute__((ext_vector_type(2))) float v2f;
typedef __attribute__((ext_vector_type(4))) float v4f;
typedef __attribute__((ext_vector_type(8))) float v8f;

#define BATCH 32
#define C_CH  256
#define CR    16      // C / R
#define HW    16384   // 128*128

// ---------------------------------------------------------------------------
// Kernel 1: squeeze — per-(b,c) spatial mean. One block per plane.
// 256 threads * 16 x b128 loads = 16384 elements, fully coalesced.
// ---------------------------------------------------------------------------
__global__ void se_mean_kernel(const float* __restrict__ x, float* __restrict__ y) {
    __shared__ float red[256];
    const int plane = blockIdx.x;  // b*256 + c
    const v4f* xp = (const v4f*)(x + (size_t)plane * HW);

    float sum = 0.f;
#pragma unroll
    for (int j = 0; j < 16; ++j) {
        v4f v = xp[j * 256 + threadIdx.x];
        sum += (v.x + v.y) + (v.z + v.w);
    }
    red[threadIdx.x] = sum;
    __syncthreads();

    if (threadIdx.x < 128) red[threadIdx.x] += red[threadIdx.x + 128];
    __syncthreads();
    if (threadIdx.x < 64)  red[threadIdx.x] += red[threadIdx.x + 64];
    __syncthreads();
    if (threadIdx.x < 32) {
        float v = red[threadIdx.x] + red[threadIdx.x + 32];
        // wave32 tree reduction
#pragma unroll
        for (int off = 16; off > 0; off >>= 1)
            v += __shfl_down(v, off, 32);
        if (threadIdx.x == 0) y[plane] = v * (1.0f / (float)HW);
    }
}

// ---------------------------------------------------------------------------
// Kernel 2: excitation MLP via V_WMMA_F32_16X16X4_F32.
// One block, 64 threads = 2 full wave32s (EXEC all-1s as WMMA requires).
// Wave w owns batch rows [16w, 16w+16).
//
// A layout (16x4 f32): lanes 0-15 -> M=lane, {K=0,K=1}; lanes 16-31 -> {K=2,K=3}
// B layout (4x16 f32): lanes 0-15 -> N=lane, {K=0,K=1}; lanes 16-31 -> {K=2,K=3}
// C/D layout (16x16 f32): VGPR r: lanes 0-15 -> M=r, N=lane;
//                                  lanes 16-31 -> M=r+8, N=lane-16
// ---------------------------------------------------------------------------
__global__ void se_excite_kernel(const float* __restrict__ y,
                                 const float* __restrict__ w1,
                                 const float* __restrict__ w2,
                                 float* __restrict__ s) {
    __shared__ float hbuf[BATCH * CR];  // h = relu(y @ w1^T), 32x16 row-major

    const int lane  = threadIdx.x & 31;
    const int wave  = threadIdx.x >> 5;   // batch tile: 0 or 1
    const int mrow  = lane & 15;          // M row (A) / N col (B,D) in tile
    const int khalf = lane >> 4;          // 0 -> K{0,1}, 1 -> K{2,3}
    const int m     = wave * 16 + mrow;

    // --- GEMM1: D(16x16) = y_tile(16x256) @ w1^T(256x16), 64 k-steps ---
    v8f acc = {};
    for (int k0 = 0; k0 < C_CH; k0 += 4) {
        const int k = k0 + 2 * khalf;
        v2f a, b;
        a[0] = y[m * C_CH + k];
        a[1] = y[m * C_CH + k + 1];
        b[0] = w1[mrow * C_CH + k];     // B[k][n] = w1[n][k], n = mrow
        b[1] = w1[mrow * C_CH + k + 1];
        acc = __builtin_amdgcn_wmma_f32_16x16x4_f32(
            false, a, false, b, (short)0, acc, false, false);
    }

    // ReLU, stage h (32x16) to LDS in row-major for GEMM2 A-operand reload
#pragma unroll
    for (int r = 0; r < 8; ++r) {
        const int dm = wave * 16 + (khalf ? r + 8 : r);
        hbuf[dm * CR + mrow] = fmaxf(acc[r], 0.f);
    }
    __syncthreads();

    // --- GEMM2: s(32x256) = h(32x16) @ w2^T(16x256), 16 N-tiles, 4 k-steps ---
    for (int nt = 0; nt < 16; ++nt) {
        v8f acc2 = {};
#pragma unroll
        for (int k0 = 0; k0 < CR; k0 += 4) {
            const int k = k0 + 2 * khalf;
            const int n = nt * 16 + mrow;
            v2f a, b;
            a[0] = hbuf[m * CR + k];
            a[1] = hbuf[m * CR + k + 1];
            b[0] = w2[n * CR + k];      // B[k][n] = w2[n][k]
            b[1] = w2[n * CR + k + 1];
            acc2 = __builtin_amdgcn_wmma_f32_16x16x4_f32(
                false, a, false, b, (short)0, acc2, false, false);
        }
#pragma unroll
        for (int r = 0; r < 8; ++r) {
            const int dm = wave * 16 + (khalf ? r + 8 : r);
            const int dn = nt * 16 + mrow;
            s[dm * C_CH + dn] = 1.0f / (1.0f + __expf(-acc2[r]));
        }
    }
}

// ---------------------------------------------------------------------------
// Kernel 3: broadcast scale. 4 blocks per (b,c) plane; scale is block-uniform
// (scalar load). b128 loads, non-temporal b128 stores (output never re-read —
// keep it out of L2).
// ---------------------------------------------------------------------------
__global__ void se_scale_kernel(const float* __restrict__ x,
                                const float* __restrict__ s,
                                float* __restrict__ out) {
    const int plane   = blockIdx.x >> 2;  // b*256 + c
    const float scale = s[plane];
    const v4f* xp = (const v4f*)x + (size_t)blockIdx.x * 1024;
    v4f*       op = (v4f*)out     + (size_t)blockIdx.x * 1024;
#pragma unroll
    for (int j = 0; j < 4; ++j) {
        v4f v = xp[j * 256 + threadIdx.x];
        v = v * scale;
        __builtin_nontemporal_store(v, op + j * 256 + threadIdx.x);
    }
}

// ---------------------------------------------------------------------------
extern "C" void kernel_launch(void* const* d_in, const int* in_sizes, int n_in,
                              void* d_out, int out_size, void* d_ws, size_t ws_size,
                              hipStream_t stream) {
    const float* x  = (const float*)d_in[0];   // (32, 256, 128, 128)
    const float* w1 = (const float*)d_in[1];   // (16, 256)
    const float* w2 = (const float*)d_in[2];   // (256, 16)
    float* out = (float*)d_out;

    float* y = (float*)d_ws;          // 32*256 means
    float* s = y + BATCH * C_CH;      // 32*256 sigmoid scales

    se_mean_kernel <<<BATCH * C_CH,     256, 0, stream>>>(x, y);
    se_excite_kernel<<<1,                64, 0, stream>>>(y, w1, w2, s);
    se_scale_kernel <<<BATCH * C_CH * 4, 256, 0, stream>>>(x, s, out);
}